// Assigner_111669150292
// MI455X (gfx1250) — compile-verified
//
#include <hip/hip_runtime.h>
#include <hip/hip_bf16.h>
#include <math.h>

// Problem constants (fixed by the reference setup)
#define B_   16
#define A_   8400
#define NC_  80
#define M_   64
#define TK_  13
#define EPS_ 1e-9f

typedef __attribute__((ext_vector_type(2))) float v2f;
typedef __attribute__((ext_vector_type(8))) float v8f;

// ---------- CDNA5 async global->LDS staging (ASYNCcnt path) ----------
static __device__ __forceinline__ void async_g2l_b32(void* lds, const void* g) {
    asm volatile("global_load_async_to_lds_b32 %0, %1, off"
                 :: "v"((unsigned)(unsigned long long)(lds)),
                    "v"((unsigned long long)(g))
                 : "memory");
}
static __device__ __forceinline__ void wait_async0() {
    asm volatile("s_wait_asynccnt 0" ::: "memory");
}

// ---------- CIoU exactly matching the reference ----------
static __device__ __forceinline__ float ciou_f(
    float gx1, float gy1, float gx2, float gy2,
    float px1, float py1, float px2, float py2)
{
    const float e = 1e-7f;
    float w1 = gx2 - gx1, h1 = gy2 - gy1 + e;   // box1 = gt
    float w2 = px2 - px1, h2 = py2 - py1 + e;   // box2 = pd
    float iw = fmaxf(fminf(gx2, px2) - fmaxf(gx1, px1), 0.f);
    float ih = fmaxf(fminf(gy2, py2) - fmaxf(gy1, py1), 0.f);
    float inter = iw * ih;
    float uni = w1 * h1 + w2 * h2 - inter + e;
    float iou = inter / uni;
    float cw = fmaxf(gx2, px2) - fminf(gx1, px1);
    float ch = fmaxf(gy2, py2) - fminf(gy1, py1);
    float c2 = cw * cw + ch * ch + e;
    float dx = px1 + px2 - gx1 - gx2;
    float dy = py1 + py2 - gy1 - gy2;
    float rho2 = (dx * dx + dy * dy) * 0.25f;
    float dat = atanf(w2 / h2) - atanf(w1 / h1);
    const float c4pi2 = 0.405284734569351f;   // 4/pi^2
    float v = c4pi2 * dat * dat;
    float alpha = v / (v - iou + (1.0f + e)); // stop_gradient irrelevant (fwd only)
    return iou - (rho2 / c2 + v * alpha);
}

// ---------- Stage 0: zero the per-GT max accumulators ----------
__global__ void k_init(float* posAlign, float* posOv) {
    int i = blockIdx.x * 256 + threadIdx.x;
    if (i < B_ * M_) { posAlign[i] = 0.f; posOv[i] = 0.f; }
}

// ---------- Stage 1: per-(b,m) top-13 anchor indices over align metric ----------
// Entire 8400-float align row lives in LDS (33.6KB of the 320KB WGP pool).
__global__ void __launch_bounds__(256) k_topk(
    const float* __restrict__ pd_scores, const float* __restrict__ pd_bboxes,
    const float* __restrict__ anc, const int* __restrict__ gt_labels,
    const float* __restrict__ gt_bboxes, const float* __restrict__ mask_gt,
    int* __restrict__ topk_out)
{
    __shared__ float s_align[A_];
    __shared__ float s_v[256];
    __shared__ int   s_i[256];
    const int bm  = blockIdx.x;          // b*M_ + m
    const int tid = threadIdx.x;
    const int b   = bm / M_;
    const float mg  = mask_gt[bm];
    const int   lab = gt_labels[bm];
    const float gx1 = gt_bboxes[bm * 4 + 0], gy1 = gt_bboxes[bm * 4 + 1];
    const float gx2 = gt_bboxes[bm * 4 + 2], gy2 = gt_bboxes[bm * 4 + 3];
    const bool valid = (mg > 0.f);

    for (int a = tid; a < A_; a += 256) {
        float al = 0.f;
        if (valid) {
            float ax = anc[a * 2 + 0], ay = anc[a * 2 + 1];
            float dmin = fminf(fminf(ax - gx1, ay - gy1), fminf(gx2 - ax, gy2 - ay));
            if (dmin > EPS_) {
                const float* pb = pd_bboxes + ((size_t)b * A_ + a) * 4;
                float iou = ciou_f(gx1, gy1, gx2, gy2, pb[0], pb[1], pb[2], pb[3]);
                float ov  = fmaxf(iou, 0.f);
                float s   = pd_scores[((size_t)b * A_ + a) * NC_ + lab];
                float o2  = ov * ov;
                al = s * o2 * o2 * o2;   // score^1 * ov^6 (ALPHA=1, BETA=6)
            }
        }
        s_align[a] = al;
    }
    __syncthreads();

    if (!valid) {   // reference: masked rows -> indices collapse to 0 -> counts>1 -> dropped
        if (tid < TK_) topk_out[bm * TK_ + tid] = -1;
        return;
    }

    for (int it = 0; it < TK_; ++it) {
        float bv = -1.f; int bi = 0;
        for (int a = tid; a < A_; a += 256) {    // strided scan keeps lowest index on ties
            float v = s_align[a];
            if (v > bv) { bv = v; bi = a; }
        }
        s_v[tid] = bv; s_i[tid] = bi;
        __syncthreads();
        for (int s = 128; s > 0; s >>= 1) {
            if (tid < s) {
                float v2 = s_v[tid + s]; int i2 = s_i[tid + s];
                if (v2 > s_v[tid] || (v2 == s_v[tid] && i2 < s_i[tid])) {
                    s_v[tid] = v2; s_i[tid] = i2;
                }
            }
            __syncthreads();
        }
        if (tid == 0) {
            int win = s_i[0];
            topk_out[bm * TK_ + it] = win;
            s_align[win] = -1.f;     // exclude; align >= 0 everywhere
        }
        __syncthreads();
    }
}

// ---------- Stage 2: per-(b,a) assignment + conflict resolution ----------
__global__ void __launch_bounds__(256) k_assign(
    const float* __restrict__ pd_scores, const float* __restrict__ pd_bboxes,
    const float* __restrict__ anc, const int* __restrict__ gt_labels,
    const float* __restrict__ gt_bboxes, const float* __restrict__ mask_gt,
    const int* __restrict__ topk, int* __restrict__ tgt,
    float* __restrict__ alignAt, float* __restrict__ posAlign,
    float* __restrict__ posOv, float* __restrict__ out_bboxes,
    float* __restrict__ out_fg)
{
    __shared__ float s_gt[M_ * 4];
    __shared__ float s_mg[M_];
    __shared__ int   s_lab[M_];
    __shared__ int   s_tk[M_ * TK_];
    const int b   = blockIdx.y;
    const int tid = threadIdx.x;

    // Async-stage the full per-batch GT descriptor set into LDS
    for (int i = tid; i < M_ * 4;   i += 256) async_g2l_b32(&s_gt[i],  &gt_bboxes[(size_t)b * M_ * 4 + i]);
    for (int i = tid; i < M_;       i += 256) async_g2l_b32(&s_mg[i],  &mask_gt[b * M_ + i]);
    for (int i = tid; i < M_;       i += 256) async_g2l_b32(&s_lab[i], &gt_labels[b * M_ + i]);
    for (int i = tid; i < M_ * TK_; i += 256) async_g2l_b32(&s_tk[i],  &topk[(size_t)b * M_ * TK_ + i]);
    wait_async0();
    __syncthreads();

    const int a = blockIdx.x * 256 + tid;
    if (a >= A_) return;

    const float ax = anc[a * 2 + 0], ay = anc[a * 2 + 1];
    const float* pb = pd_bboxes + ((size_t)b * A_ + a) * 4;
    const float p0 = pb[0], p1 = pb[1], p2 = pb[2], p3 = pb[3];
    const float* srow = pd_scores + ((size_t)b * A_ + a) * NC_;

    int   fg0 = 0, firstM0 = 0;
    float bestOv = -1.f; int bestM = 0;      // argmax over overlaps column, first-index ties
    for (int m = 0; m < M_; ++m) {
        float mg  = s_mg[m];
        float gx1 = s_gt[m * 4 + 0], gy1 = s_gt[m * 4 + 1];
        float gx2 = s_gt[m * 4 + 2], gy2 = s_gt[m * 4 + 3];
        float ov = 0.f; bool gm = false;
        if (mg > 0.f) {
            float dmin = fminf(fminf(ax - gx1, ay - gy1), fminf(gx2 - ax, gy2 - ay));
            if (dmin > EPS_) {
                gm = true;
                ov = fmaxf(ciou_f(gx1, gy1, gx2, gy2, p0, p1, p2, p3), 0.f);
            }
        }
        if (ov > bestOv) { bestOv = ov; bestM = m; }
        if (gm) {
            bool intk = false;
#pragma unroll
            for (int t = 0; t < TK_; ++t) intk |= (s_tk[m * TK_ + t] == a);
            if (intk) { if (fg0 == 0) firstM0 = m; fg0++; }
        }
    }
    int sel; float fg;
    if      (fg0 > 1)  { sel = bestM;   fg = 1.f; }   // multi-assigned -> argmax overlap
    else if (fg0 == 1) { sel = firstM0; fg = 1.f; }
    else               { sel = 0;       fg = 0.f; }   // argmax of all-zero mask -> 0

    // Recompute metrics only at the selected GT (cheap; avoids B*M*A spill to HBM)
    float alS = 0.f, ovS = 0.f;
    {
        float mg  = s_mg[sel];
        float gx1 = s_gt[sel * 4 + 0], gy1 = s_gt[sel * 4 + 1];
        float gx2 = s_gt[sel * 4 + 2], gy2 = s_gt[sel * 4 + 3];
        if (mg > 0.f) {
            float dmin = fminf(fminf(ax - gx1, ay - gy1), fminf(gx2 - ax, gy2 - ay));
            if (dmin > EPS_) {
                ovS = fmaxf(ciou_f(gx1, gy1, gx2, gy2, p0, p1, p2, p3), 0.f);
                float s  = srow[s_lab[sel]];
                float o2 = ovS * ovS;
                alS = s * o2 * o2 * o2;
            }
        }
    }
    const size_t ba = (size_t)b * A_ + a;
    tgt[ba]     = sel;
    alignAt[ba] = alS;
    out_fg[ba]  = fg;
    out_bboxes[ba * 4 + 0] = s_gt[sel * 4 + 0];
    out_bboxes[ba * 4 + 1] = s_gt[sel * 4 + 1];
    out_bboxes[ba * 4 + 2] = s_gt[sel * 4 + 2];
    out_bboxes[ba * 4 + 3] = s_gt[sel * 4 + 3];
    if (fg > 0.f) {   // values >= 0: uint ordering == float ordering
        atomicMax((unsigned int*)&posAlign[b * M_ + sel], __float_as_uint(alS));
        atomicMax((unsigned int*)&posOv[b * M_ + sel],   __float_as_uint(ovS));
    }
}

// ---------- Stage 3: target_scores via WMMA ----------
// target_scores = S (A x 64) * H (64 x 80), S has one norm value per fg row,
// H = one-hot(labels). Emitted as f32 16x16x4 WMMA tiles (A=525*16, NC=5*16, K=16*4).
__global__ void __launch_bounds__(160) k_scores(
    const int* __restrict__ gt_labels, const int* __restrict__ tgt,
    const float* __restrict__ alignAt, const float* __restrict__ posAlign,
    const float* __restrict__ posOv, const float* __restrict__ fgm,
    float* __restrict__ out_scores)
{
    __shared__ int s_lab[M_];
    const int tid  = threadIdx.x;
    const int b    = blockIdx.x / (A_ / 16);
    const int tile = blockIdx.x % (A_ / 16);

    if (tid < M_) async_g2l_b32(&s_lab[tid], &gt_labels[b * M_ + tid]);
    wait_async0();
    __syncthreads();

    const int lane = tid & 31;
    const int ct   = tid >> 5;           // 5 waves -> 5 class tiles of 16
    const int base = tile * 16;
    const int arow = base + (lane & 15);
    const size_t ba = (size_t)b * A_ + arow;

    const float fg  = fgm[ba];
    const int   sel = tgt[ba];
    float val = 0.f;
    if (fg > 0.f)
        val = alignAt[ba] * posOv[b * M_ + sel] / (posAlign[b * M_ + sel] + EPS_);

    const int cc    = ct * 16 + (lane & 15);
    const int khalf = (lane < 16) ? 0 : 2;   // 16x4 f32 A/B lane->K mapping

    v8f acc = {0.f, 0.f, 0.f, 0.f, 0.f, 0.f, 0.f, 0.f};
#pragma unroll
    for (int kk = 0; kk < M_ / 4; ++kk) {
        int mk = kk * 4 + khalf;
        v2f Af, Bf;
        Af.x = (sel == mk    ) ? val : 0.f;
        Af.y = (sel == mk + 1) ? val : 0.f;
        Bf.x = (s_lab[mk    ] == cc) ? 1.f : 0.f;
        Bf.y = (s_lab[mk + 1] == cc) ? 1.f : 0.f;
        acc = __builtin_amdgcn_wmma_f32_16x16x4_f32(
            false, Af, false, Bf, (short)0, acc, false, false);
    }
#pragma unroll
    for (int r = 0; r < 8; ++r) {
        int row = (lane < 16) ? r : (r + 8);   // 16x16 f32 D layout
        out_scores[((size_t)b * A_ + base + row) * NC_ + cc] = acc[r];
    }
}

extern "C" void kernel_launch(void* const* d_in, const int* in_sizes, int n_in,
                              void* d_out, int out_size, void* d_ws, size_t ws_size,
                              hipStream_t stream) {
    (void)in_sizes; (void)n_in; (void)out_size; (void)ws_size;
    const float* pd_scores = (const float*)d_in[0];
    const float* pd_bboxes = (const float*)d_in[1];
    const float* anc       = (const float*)d_in[2];
    const int*   gt_labels = (const int*)  d_in[3];
    const float* gt_bboxes = (const float*)d_in[4];
    const float* mask_gt   = (const float*)d_in[5];

    char* ws = (char*)d_ws;
    int*   topk     = (int*)ws;                                                   // B*M*TK
    int*   tgt      = (int*)(ws + (size_t)B_ * M_ * TK_ * 4);                     // B*A
    float* alignAt  = (float*)(ws + (size_t)B_ * M_ * TK_ * 4 + (size_t)B_ * A_ * 4);
    float* posAlign = (float*)(ws + (size_t)B_ * M_ * TK_ * 4 + (size_t)B_ * A_ * 8);
    float* posOv    = posAlign + B_ * M_;

    float* out_bboxes = (float*)d_out;                       // B*A*4
    float* out_scores = out_bboxes + (size_t)B_ * A_ * 4;    // B*A*NC
    float* out_fg     = out_scores + (size_t)B_ * A_ * NC_;  // B*A

    k_init<<<(B_ * M_ + 255) / 256, 256, 0, stream>>>(posAlign, posOv);
    k_topk<<<B_ * M_, 256, 0, stream>>>(pd_scores, pd_bboxes, anc,
                                        gt_labels, gt_bboxes, mask_gt, topk);
    dim3 g2((A_ + 255) / 256, B_);
    k_assign<<<g2, 256, 0, stream>>>(pd_scores, pd_bboxes, anc, gt_labels, gt_bboxes,
                                     mask_gt, topk, tgt, alignAt, posAlign, posOv,
                                     out_bboxes, out_fg);
    k_scores<<<B_ * (A_ / 16), 160, 0, stream>>>(gt_labels, tgt, alignAt,
                                                 posAlign, posOv, out_fg, out_scores);
}